// MaskingActivationLayer_74182675136606
// MI455X (gfx1250) — compile-verified
//
#include <hip/hip_runtime.h>
#include <cstdint>

#define SEQ 2047
#define NB  8
#define RT  1162
#define RT2 581          // RT/2, exact (1162 dwords = 581 x b64)
#define K1T 256
#define TPT 8            // ceil(2047/256)

// ---------------------------------------------------------------------------
// Kernel 1: per-batch prefix scan of song state.
// State per position j (inclusive over tokens 0..j):
//   - 129-bit instrument-present bitmask (5 u32, OR-scan)
//   - last index of type 4/5/6 tokens (max-scan, default 0 -> gathers song[b,0,f])
//   - has_ks / has_ts flags (OR-scan)
// Stored as 6 u32 per position: m0..m4, packed(ksval|tsval<<8|tpval<<16|flags<<24)
// ---------------------------------------------------------------------------
__global__ __launch_bounds__(256)
void prefix_scan_kernel(const int* __restrict__ song, uint32_t* __restrict__ ws) {
  __shared__ uint32_t sm0[K1T], sm1[K1T], sm2[K1T], sm3[K1T], sm4[K1T], sfl[K1T];
  __shared__ int sk[K1T], st[K1T], sp2[K1T];
  const int b = blockIdx.x;
  const int tid = threadIdx.x;
  const int* sb = song + (size_t)b * SEQ * 11;
  const int j0 = tid * TPT;

  uint32_t m0=0,m1=0,m2=0,m3=0,m4=0,fl=0;
  int ki=0, ti=0, pi=0;
  for (int k = 0; k < TPT; ++k) {
    int j = j0 + k;
    if (j < SEQ) {
      int ty = sb[j*11];
      if (ty == 1) {
        int v = sb[j*11 + 6];
        uint32_t bit = 1u << (v & 31);
        int w = v >> 5;
        if (w==0) m0|=bit; else if (w==1) m1|=bit; else if (w==2) m2|=bit;
        else if (w==3) m3|=bit; else m4|=bit;
      } else if (ty == 4) { ki = j; fl |= 1u; }
      else if (ty == 5)   { ti = j; fl |= 2u; }
      else if (ty == 6)   { pi = j; }
    }
  }
  sm0[tid]=m0; sm1[tid]=m1; sm2[tid]=m2; sm3[tid]=m3; sm4[tid]=m4;
  sfl[tid]=fl; sk[tid]=ki; st[tid]=ti; sp2[tid]=pi;
  __syncthreads();

  // Hillis-Steele inclusive scan over the 256 thread aggregates
  for (int s = 1; s < K1T; s <<= 1) {
    uint32_t a0=sm0[tid],a1=sm1[tid],a2=sm2[tid],a3=sm3[tid],a4=sm4[tid],af=sfl[tid];
    int ak=sk[tid], at=st[tid], ap=sp2[tid];
    if (tid >= s) {
      a0|=sm0[tid-s]; a1|=sm1[tid-s]; a2|=sm2[tid-s]; a3|=sm3[tid-s]; a4|=sm4[tid-s];
      af|=sfl[tid-s];
      ak = max(ak, sk[tid-s]); at = max(at, st[tid-s]); ap = max(ap, sp2[tid-s]);
    }
    __syncthreads();
    sm0[tid]=a0; sm1[tid]=a1; sm2[tid]=a2; sm3[tid]=a3; sm4[tid]=a4;
    sfl[tid]=af; sk[tid]=ak; st[tid]=at; sp2[tid]=ap;
    __syncthreads();
  }

  // exclusive prefix for this thread
  uint32_t e0=0,e1=0,e2=0,e3=0,e4=0,ef=0; int ek=0,et=0,ep=0;
  if (tid > 0) {
    e0=sm0[tid-1]; e1=sm1[tid-1]; e2=sm2[tid-1]; e3=sm3[tid-1]; e4=sm4[tid-1];
    ef=sfl[tid-1]; ek=sk[tid-1]; et=st[tid-1]; ep=sp2[tid-1];
  }
  for (int k = 0; k < TPT; ++k) {
    int j = j0 + k;
    if (j >= SEQ) break;
    int ty = sb[j*11];
    if (ty == 1) {
      int v = sb[j*11+6];
      uint32_t bit = 1u << (v & 31);
      int w = v >> 5;
      if (w==0) e0|=bit; else if (w==1) e1|=bit; else if (w==2) e2|=bit;
      else if (w==3) e3|=bit; else e4|=bit;
    } else if (ty == 4) { ek = j; ef |= 1u; }
    else if (ty == 5)   { et = j; ef |= 2u; }
    else if (ty == 6)   { ep = j; }
    uint32_t ksv = (uint32_t)sb[ek*11 + 8]  & 0xFFu;
    uint32_t tsv = (uint32_t)sb[et*11 + 9]  & 0xFFu;
    uint32_t tpv = (uint32_t)sb[ep*11 + 10] & 0xFFu;
    uint32_t packed = ksv | (tsv<<8) | (tpv<<16) | ((ef&3u)<<24);
    uint32_t* wp = ws + (size_t)(b*SEQ + j) * 6;
    wp[0]=e0; wp[1]=e1; wp[2]=e2; wp[3]=e3; wp[4]=e4; wp[5]=packed;
  }
}

// ---------------------------------------------------------------------------
// Kernel 2: one block per (b,i) row. Async-stage the 1162-float row into LDS
// with gfx1250 global_load_async_to_lds_b64 (ASYNCcnt), argmax from LDS,
// then region-masked f32x2 write-out. Single HBM read + single HBM write.
// ---------------------------------------------------------------------------
__global__ __launch_bounds__(256)
void mask_kernel(const int* __restrict__ song, const int* __restrict__ ctype,
                 const float* __restrict__ scores, const uint32_t* __restrict__ ws,
                 float* __restrict__ out) {
  __shared__ __align__(16) float row[RT];
  __shared__ float rv[256];
  __shared__ int   ri[256];
  __shared__ int   s_pred[2];

  const int blk = blockIdx.x;
  const int b = blk / SEQ;
  const int i = blk - b * SEQ;
  const int tid = threadIdx.x;
  const size_t base = (size_t)blk * RT;      // dword offset; byte offset = blk*4648 (8B aligned)
  const float* src = scores + base;

  // async copy row -> LDS, 8 bytes per lane per instruction (581 pairs exactly)
  for (int p = tid; p < RT2; p += 256) {
    uint32_t laddr = (uint32_t)(uintptr_t)(&row[2*p]);
    unsigned long long ga = (unsigned long long)(uintptr_t)(src + 2*p);
    asm volatile("global_load_async_to_lds_b64 %0, %1, off"
                 :: "v"(laddr), "v"(ga) : "memory");
  }
  asm volatile("s_wait_asynccnt 0" ::: "memory");
  __syncthreads();

  // argmax over scores[0:256] (first-index tie-break, like jnp.argmax)
  rv[tid] = row[tid]; ri[tid] = tid;
  __syncthreads();
  for (int s = 128; s > 0; s >>= 1) {
    if (tid < s) {
      float va = rv[tid], vb = rv[tid+s];
      int   ia = ri[tid], ib = ri[tid+s];
      if (vb > va || (vb == va && ib < ia)) { rv[tid] = vb; ri[tid] = ib; }
    }
    __syncthreads();
  }
  if (tid == 0) {
    s_pred[0] = ri[0];
    float bv = row[256]; int bi = 0;
    for (int k = 1; k < 16; ++k) { float v = row[256+k]; if (v > bv) { bv = v; bi = k; } }
    s_pred[1] = bi;
  }
  __syncthreads();

  const int row_idx = b * SEQ + i;
  const int* tok = song + (size_t)row_idx * 11;
  const int t    = ctype[row_idx];
  const int minm = tok[1], sbv = tok[2], spv = tok[3];
  const int jp = (i + 1 < SEQ) ? (i + 1) : (SEQ - 1);   // _prefix_gather shift
  const uint32_t* wp = ws + (size_t)(b*SEQ + jp) * 6;
  const uint32_t im0=wp[0], im1=wp[1], im2=wp[2], im3=wp[3], im4=wp[4], pk=wp[5];
  const int ksv = pk & 0xFF, tsv = (pk>>8)&0xFF, tpv = (pk>>16)&0xFF;
  const bool hasks = (pk >> 24) & 1u, hasts = (pk >> 25) & 1u;
  const bool hasinst = (im0|im1|im2|im3|im4) != 0u;

  const bool c1 = (s_pred[0] == minm);
  const int  minbeat = c1 ? sbv : 0;
  const bool c2 = c1 && (s_pred[1] == minbeat);
  const int  minpos = c2 ? spv : 0;

  // NUMERATORS = {1,2,3,4,5,6,7,8,9,10,12,16} packed 5 bits each
  const unsigned long long NUMTAB =
      (1ULL<<0)|(2ULL<<5)|(3ULL<<10)|(4ULL<<15)|(5ULL<<20)|(6ULL<<25)|
      (7ULL<<30)|(8ULL<<35)|(9ULL<<40)|(10ULL<<45)|(12ULL<<50)|(16ULL<<55);
  const int maxbeat = (int)((NUMTAB >> (5 * (tsv % 12))) & 31ULL);
  const int minm46  = (sbv == 0 && spv == 0) ? minm : minm + 1;
  const bool is3  = (t == 3);
  const bool is46 = (t >= 4) && (t <= 6);

  auto keep_at = [&](int r) -> bool {
    bool keep = true;
    if (r < 256) {                       // measure
      if (is3)       keep = (r >= minm);
      else if (is46) keep = (r >= minm46);
    } else if (r < 272) {                // beat
      if (is3) { int v = r - 256; keep = (v >= minbeat) && (v < maxbeat); }
    } else if (r < 400) {                // position
      if (is3) keep = ((r - 272) >= minpos);
    } else if (r < 784) {                // duration + pitch: always true
    } else if (r < 913) {                // instrument
      int v = r - 784;
      uint32_t wsel = (v < 32) ? im0 : (v < 64) ? im1 : (v < 96) ? im2
                      : (v < 128) ? im3 : im4;
      bool bit = (wsel >> (v & 31)) & 1u;
      if (is3)          keep = bit;
      else if (t == 1)  keep = hasinst ? !bit : true;
    } else if (r < 1041) {               // velocity: always true
    } else if (r < 1065) {               // key_sign
      int v = r - 1041;
      if (is3)         keep = (v == ksv);
      else if (t == 4) keep = hasks ? (v != ksv) : true;
    } else if (r < 1113) {               // time_sign
      int v = r - 1065;
      if (is3)         keep = (v == tsv);
      else if (t == 5) keep = hasts ? (v != tsv) : true;
    } else {                             // tempo
      int v = r - 1113;
      if (is3)         keep = (v == tpv);
      else if (t == 6) keep = (v != tpv);
    }
    return keep;
  };

  for (int p = tid; p < RT2; p += 256) {
    int r = 2 * p;
    float2 o;
    o.x = keep_at(r)     ? row[r]     : -1e9f;
    o.y = keep_at(r + 1) ? row[r + 1] : -1e9f;
    *(float2*)(out + base + r) = o;    // 8B-aligned b64 store
  }
}

extern "C" void kernel_launch(void* const* d_in, const int* in_sizes, int n_in,
                              void* d_out, int out_size, void* d_ws, size_t ws_size,
                              hipStream_t stream) {
  (void)in_sizes; (void)n_in; (void)out_size; (void)ws_size;
  const int*   song   = (const int*)d_in[0];
  const int*   ctype  = (const int*)d_in[1];
  const float* scores = (const float*)d_in[2];
  float*       out    = (float*)d_out;
  uint32_t*    ws     = (uint32_t*)d_ws;   // needs 8*2047*6*4 = 393,024 bytes

  prefix_scan_kernel<<<NB, 256, 0, stream>>>(song, ws);
  mask_kernel<<<NB * SEQ, 256, 0, stream>>>(song, ctype, scores, ws, out);
}